// PerfeCT_11141145166215
// MI455X (gfx1250) — compile-verified
//
#include <hip/hip_runtime.h>
#include <stdint.h>

// ---------------------------------------------------------------------------
// Exact set membership of int64 triple keys via an open-addressing hash table
// kept resident in MI455X's 192 MB L2.
//
//   key(h,r,t) = (h*15000 + r)*15000 + t   < 15000^3 ~ 3.4e12  (fits i64)
//   table: 2^24 slots * 8B = 134 MB  -> fits 192 MB L2; LF = 10M/16.7M = 0.60
//   sentinel EMPTY = 0xFFFF..F (never a valid key)
//
// Phase 1: fill table with EMPTY (128-bit stores)
// Phase 2: stream data[3][N] with non-temporal b128 loads (2 triples/thread;
//          NT protects the L2-resident table from eviction), insert keys
//          with 64-bit global CAS
// Phase 3: probe 100K queries against the L2-resident table, write
//          10*(member-0.5) = +/-5.0f
// ---------------------------------------------------------------------------

#define EMPTY_KEY 0xFFFFFFFFFFFFFFFFull
#define N_ENT_LL  15000ll

typedef unsigned long long ull;
typedef long long          sll;
typedef sll sll2 __attribute__((ext_vector_type(2)));   // NT-loadable vector

__device__ __forceinline__ ull mix64(ull x) {
    // splitmix64 finalizer: full-avalanche 64-bit hash
    x += 0x9E3779B97F4A7C15ull;
    x = (x ^ (x >> 30)) * 0xBF58476D1CE4E5B9ull;
    x = (x ^ (x >> 27)) * 0x94D049BB133111EBull;
    return x ^ (x >> 31);
}

__device__ __forceinline__ void insert_key(ull* __restrict__ table, ull mask, ull key) {
    ull slot = mix64(key) & mask;
    // linear probing insert; a key already present ends the probe, so duplicates are stored once
    for (ull p = 0; p <= mask; ++p) {
        ull prev = atomicCAS(&table[slot], EMPTY_KEY, key); // global_atomic_cmpswap_b64
        if (prev == EMPTY_KEY || prev == key) return;
        slot = (slot + 1) & mask;
    }
}

// ---- Phase 1: table init ---------------------------------------------------
__global__ void perfect_init_table(ull* __restrict__ table, ull n_slots) {
    ull i      = (ull)blockIdx.x * blockDim.x + threadIdx.x;
    ull stride = (ull)gridDim.x * blockDim.x;
    ulonglong2* t2 = reinterpret_cast<ulonglong2*>(table);
    const ulonglong2 v = make_ulonglong2(EMPTY_KEY, EMPTY_KEY);
    const ull n2 = n_slots >> 1;                       // n_slots is a power of two
    for (ull j = i; j < n2; j += stride)
        t2[j] = v;                                      // global_store_b128
}

// ---- Phase 2a: streaming insert, 2 triples per thread (N even) -------------
__global__ void perfect_insert_vec(const sll2* __restrict__ v0,
                                   const sll2* __restrict__ v1,
                                   const sll2* __restrict__ v2,
                                   sll npairs,
                                   ull* __restrict__ table, ull mask) {
    sll i = (sll)blockIdx.x * blockDim.x + threadIdx.x;
    const sll stride = (sll)gridDim.x * blockDim.x;
    for (; i < npairs; i += stride) {
        if (i + stride < npairs) {                      // global_prefetch_b8
            __builtin_prefetch(&v0[i + stride], 0, 0);
            __builtin_prefetch(&v1[i + stride], 0, 0);
            __builtin_prefetch(&v2[i + stride], 0, 0);
        }
        // one-shot 240 MB stream: b128 non-temporal loads (th:NT) so the
        // L2-resident hash table is not evicted
        sll2 h = __builtin_nontemporal_load(&v0[i]);
        sll2 r = __builtin_nontemporal_load(&v1[i]);
        sll2 t = __builtin_nontemporal_load(&v2[i]);
        ull k0 = (ull)((h.x * N_ENT_LL + r.x) * N_ENT_LL + t.x);
        ull k1 = (ull)((h.y * N_ENT_LL + r.y) * N_ENT_LL + t.y);
        insert_key(table, mask, k0);
        insert_key(table, mask, k1);
    }
}

// ---- Phase 2b: scalar fallback (odd N keeps vector bases misaligned) -------
__global__ void perfect_insert_scalar(const sll* __restrict__ d0,
                                      const sll* __restrict__ d1,
                                      const sll* __restrict__ d2,
                                      sll n,
                                      ull* __restrict__ table, ull mask) {
    sll i = (sll)blockIdx.x * blockDim.x + threadIdx.x;
    const sll stride = (sll)gridDim.x * blockDim.x;
    for (; i < n; i += stride) {
        sll h = __builtin_nontemporal_load(&d0[i]);
        sll r = __builtin_nontemporal_load(&d1[i]);
        sll t = __builtin_nontemporal_load(&d2[i]);
        insert_key(table, mask, (ull)((h * N_ENT_LL + r) * N_ENT_LL + t));
    }
}

// ---- Phase 3: probe --------------------------------------------------------
__global__ void perfect_probe(const sll* __restrict__ heads,
                              const sll* __restrict__ rels,
                              const sll* __restrict__ tails,
                              int q,
                              const ull* __restrict__ table, ull mask,
                              float* __restrict__ out) {
    int i = blockIdx.x * blockDim.x + threadIdx.x;
    if (i >= q) return;
    sll h = heads[i];
    sll r = rels[i];
    sll t = tails[i];
    ull key  = (ull)((h * N_ENT_LL + r) * N_ENT_LL + t);
    ull slot = mix64(key) & mask;
    float res = -5.0f;                                  // 10*(0-0.5)
    for (ull p = 0; p <= mask; ++p) {
        ull v = table[slot];                            // L2-resident read
        if (v == key)       { res = 5.0f; break; }      // 10*(1-0.5)
        if (v == EMPTY_KEY) { break; }
        slot = (slot + 1) & mask;
    }
    out[i] = res;
}

// ---------------------------------------------------------------------------
extern "C" void kernel_launch(void* const* d_in, const int* in_sizes, int n_in,
                              void* d_out, int out_size, void* d_ws, size_t ws_size,
                              hipStream_t stream) {
    const sll* heads = (const sll*)d_in[0];
    const sll* rels  = (const sll*)d_in[1];
    const sll* tails = (const sll*)d_in[2];
    const sll* data  = (const sll*)d_in[3];

    const sll Q = in_sizes[0];
    const sll N = (sll)in_sizes[3] / 3;   // data is [3, N] row-major
    const sll* d0 = data;
    const sll* d1 = data + N;
    const sll* d2 = data + 2 * N;

    // Table: largest power of two that fits in ws, capped at 2^24 slots
    // (134 MB) so the whole table stays resident in the 192 MB L2.
    ull slots_avail = (ull)(ws_size / 8);
    ull T = 2;
    while ((T << 1) <= slots_avail && T < (1ull << 24)) T <<= 1;
    const ull mask = T - 1;
    ull* table = (ull*)d_ws;

    // Phase 1: init slots with b128 stores.
    perfect_init_table<<<2048, 256, 0, stream>>>(table, T);

    // Phase 2: stream-insert N triples.
    if ((N & 1) == 0) {
        // N even -> d1/d2 bases stay 16B-aligned; b128 NT path, 2 triples/thread
        perfect_insert_vec<<<4096, 256, 0, stream>>>(
            (const sll2*)d0, (const sll2*)d1, (const sll2*)d2,
            N >> 1, table, mask);
    } else {
        perfect_insert_scalar<<<4096, 256, 0, stream>>>(d0, d1, d2, N, table, mask);
    }

    // Phase 3: Q probes.
    int q_blocks = (int)((Q + 255) / 256);
    if (q_blocks < 1) q_blocks = 1;
    perfect_probe<<<q_blocks, 256, 0, stream>>>(heads, rels, tails, (int)Q,
                                                table, mask, (float*)d_out);
}